// BertCrf_84189948936829
// MI455X (gfx1250) — compile-verified
//
#include <hip/hip_runtime.h>

typedef __attribute__((ext_vector_type(2))) float v2f;
typedef __attribute__((ext_vector_type(8))) float v8f;

#define Bsz 64
#define Ssz 512
#define Hsz 768
#define Lsz 9

// -------------------------------------------------------------------------
// Kernel 1: feats[m, l] = sum_k hidden[m, k] * W[l, k] + bias[l]
// One wave per 16-row tile, V_WMMA_F32_16X16X4_F32 accumulation over K=768.
// B (= W transposed, padded 9->16 cols with zeros) staged in LDS.
// -------------------------------------------------------------------------
__global__ __launch_bounds__(256) void gemm_feats_kernel(
    const float* __restrict__ hidden,   // (32768, 768)
    const float* __restrict__ W,        // (9, 768)
    const float* __restrict__ bias,     // (9)
    float* __restrict__ feats)          // (32768, 9)
{
    __shared__ float Wt[Hsz * 16];      // Wt[k][n] = W[n][k], n padded to 16

    const int tid = threadIdx.x;

    // zero-pad, then fill transposed W (coalesced read of W)
    for (int i = tid; i < Hsz * 16; i += 256) Wt[i] = 0.0f;
    __syncthreads();
    for (int i = tid; i < Lsz * Hsz; i += 256) {
        int l = i / Hsz, k = i % Hsz;
        Wt[k * 16 + l] = W[i];
    }
    __syncthreads();

    const int wave = tid >> 5;
    const int lane = tid & 31;
    const int half = lane >> 4;        // 0: K0/K1 , 1: K2/K3 (A & B frag halves)
    const int mn   = lane & 15;        // M index (A) / N index (B, C/D)

    const long row_base = ((long)blockIdx.x * 8 + wave) * 16;
    const float* arow = hidden + (row_base + mn) * Hsz + 2 * half;

    v8f c = {};
    #pragma unroll 8
    for (int k = 0; k < Hsz; k += 4) {
        // A fragment: 16x4 fp32 -> v0 = K(2*half), v1 = K(2*half+1), 8B load
        v2f a = *(const v2f*)(arow + k);
        // B fragment: 4x16 fp32 from LDS, conflict-free (stride 16 dwords)
        v2f bf;
        bf.x = Wt[(k + 2 * half)     * 16 + mn];
        bf.y = Wt[(k + 2 * half + 1) * 16 + mn];
        c = __builtin_amdgcn_wmma_f32_16x16x4_f32(
                /*neg_a=*/false, a, /*neg_b=*/false, bf,
                /*c_mod=*/(short)0, c, /*reuse_a=*/false, /*reuse_b=*/false);
    }

    // C/D layout: VGPR r -> M = r + 8*half, N = mn ; keep only N < 9
    if (mn < Lsz) {
        const float bn = bias[mn];
        #pragma unroll
        for (int r = 0; r < 8; ++r) {
            long m = row_base + r + 8 * half;
            feats[m * Lsz + mn] = c[r] + bn;
        }
    }
}

// -------------------------------------------------------------------------
// Kernel 2: per-batch CRF NLL. One wave per batch; lanes 0..8 = states.
// alpha gathered across lanes via __shfl (v_readlane); exp/log on trans unit.
// -------------------------------------------------------------------------
__global__ __launch_bounds__(32) void crf_batch_kernel(
    const float* __restrict__ feats,    // (64, 512, 9)
    const float* __restrict__ startT,   // (9)
    const float* __restrict__ endT,     // (9)
    const float* __restrict__ trans,    // (9, 9)
    const int* __restrict__ labels,     // (64, 512)
    const unsigned char* __restrict__ mask, // (64, 512) bool
    float* __restrict__ nll)            // (64)
{
    const int b    = blockIdx.x;
    const int lane = threadIdx.x;
    const float* fb = feats + (long)b * Ssz * Lsz;
    const unsigned char* mb = mask + (long)b * Ssz;
    const int* lb = labels + (long)b * Ssz;

    const float LOG2E = 1.4426950408889634f;
    const float LN2   = 0.6931471805599453f;

    const int j = (lane < Lsz) ? lane : 0;  // lanes >= 9 mirror lane 0 (unused)

    float tcol[Lsz];
    #pragma unroll
    for (int i = 0; i < Lsz; ++i) tcol[i] = trans[i * Lsz + j];

    // forward scan: alpha[j] lives in lane j
    float alpha = startT[j] + fb[j];
    for (int t = 1; t < Ssz; ++t) {
        const float ft = fb[t * Lsz + j];
        const bool  mt = (mb[t] != 0);
        float v[Lsz];
        float m = -3.0e38f;
        #pragma unroll
        for (int i = 0; i < Lsz; ++i) {
            float ai = __shfl(alpha, i, 32);
            v[i] = ai + tcol[i];
            m = fmaxf(m, v[i]);
        }
        float s = 0.0f;
        #pragma unroll
        for (int i = 0; i < Lsz; ++i)
            s += __builtin_amdgcn_exp2f((v[i] - m) * LOG2E);
        float neu = m + __builtin_amdgcn_logf(s) * LN2 + ft;
        alpha = mt ? neu : alpha;
    }

    // log_den = logsumexp_j(alpha[j] + end[j])
    const float d = alpha + endT[j];
    float dv[Lsz];
    float m2 = -3.0e38f;
    #pragma unroll
    for (int i = 0; i < Lsz; ++i) {
        dv[i] = __shfl(d, i, 32);
        m2 = fmaxf(m2, dv[i]);
    }
    float s2 = 0.0f;
    #pragma unroll
    for (int i = 0; i < Lsz; ++i)
        s2 += __builtin_amdgcn_exp2f((dv[i] - m2) * LOG2E);
    const float log_den = m2 + __builtin_amdgcn_logf(s2) * LN2;

    // gold-path score: lanes strided over t
    float partial = 0.0f;
    int cnt = 0;
    for (int t = lane; t < Ssz; t += 32) {
        const bool mt = (mb[t] != 0);
        cnt += mt ? 1 : 0;
        const int lt = lb[t];
        const float emit = fb[t * Lsz + lt];
        if (t == 0) {
            partial += startT[lt] + emit;
        } else if (mt) {
            partial += trans[lb[t - 1] * Lsz + lt] + emit;
        }
    }
    #pragma unroll
    for (int off = 16; off > 0; off >>= 1) {
        partial += __shfl_down(partial, off, 32);
        cnt     += __shfl_down(cnt,     off, 32);
    }
    if (lane == 0) {
        const int last_tag = lb[cnt - 1];
        nll[b] = log_den - (partial + endT[last_tag]);
    }
}

// -------------------------------------------------------------------------
// Kernel 3: mean over 64 per-batch NLLs -> scalar
// -------------------------------------------------------------------------
__global__ __launch_bounds__(32) void reduce_mean_kernel(
    const float* __restrict__ nll, float* __restrict__ out)
{
    const int lane = threadIdx.x;
    float v = nll[lane] + nll[lane + 32];
    #pragma unroll
    for (int off = 16; off > 0; off >>= 1) v += __shfl_down(v, off, 32);
    if (lane == 0) out[0] = v * (1.0f / 64.0f);
}

extern "C" void kernel_launch(void* const* d_in, const int* in_sizes, int n_in,
                              void* d_out, int out_size, void* d_ws, size_t ws_size,
                              hipStream_t stream) {
    const float* hidden = (const float*)d_in[0];            // (64,512,768) f32
    const float* W      = (const float*)d_in[1];            // (9,768) f32
    const float* bias   = (const float*)d_in[2];            // (9) f32
    const float* startT = (const float*)d_in[3];            // (9) f32
    const float* endT   = (const float*)d_in[4];            // (9) f32
    const float* trans  = (const float*)d_in[5];            // (9,9) f32
    const int*   labels = (const int*)d_in[6];              // (64,512) i32
    const unsigned char* mask = (const unsigned char*)d_in[7]; // (64,512) bool

    float* feats = (float*)d_ws;                            // 32768*9 f32
    float* per_b = feats + (size_t)Bsz * Ssz * Lsz;         // 64 f32

    // 32768 rows / (16 rows/wave * 8 waves/block) = 256 blocks
    gemm_feats_kernel<<<256, 256, 0, stream>>>(hidden, W, bias, feats);
    crf_batch_kernel<<<Bsz, 32, 0, stream>>>(feats, startT, endT, trans,
                                             labels, mask, per_b);
    reduce_mean_kernel<<<1, 32, 0, stream>>>(per_b, (float*)d_out);
}